// GCN_68813966017111
// MI455X (gfx1250) — compile-verified
//
#include <hip/hip_runtime.h>

// 2-layer GCN (GCNConv -> ReLU -> GCNConv -> scatter_mean) for MI455X (gfx1250).
// GEMMs use V_WMMA_F32_16X16X4_F32 (fp32-exact matrix pipe); edge aggregation
// uses coalesced (edge,feat) mapping with global f32 atomics; everything is
// memory-bound and sized against 23.3 TB/s HBM + 192 MB L2 (feature matrices
// are L2-resident).

typedef __attribute__((ext_vector_type(2))) float v2f;
typedef __attribute__((ext_vector_type(8))) float v8f;

#define FDIM 64   // hidden and output feature dimension

static inline int cdiv(long long a, long long b) { return (int)((a + b - 1) / b); }

// ---------------------------------------------------------------- utility ---
__global__ void k_fill(float* __restrict__ p, float v, int n) {
  int i = blockIdx.x * blockDim.x + threadIdx.x;
  if (i < n) p[i] = v;
}

// deg[dst] += 1 per edge (deg pre-filled with 1.0 for the self loop)
__global__ void k_deg(const int* __restrict__ dst, float* __restrict__ deg, int E) {
  int e = blockIdx.x * blockDim.x + threadIdx.x;
  if (e < E) atomicAdd(&deg[dst[e]], 1.0f);
}

// dis = deg^{-1/2}, in place (deg >= 1 always due to self loops)
__global__ void k_rsqrt_inplace(float* __restrict__ d, int n) {
  int i = blockIdx.x * blockDim.x + threadIdx.x;
  if (i < n) d[i] = rsqrtf(d[i]);
}

// ------------------------------------------------------------- WMMA GEMM ---
// C[M x F] = A[M x K] @ B[K x F], all row-major fp32.
// One wave per 16x16 output tile, K unrolled in steps of 4 through
// v_wmma_f32_16x16x4_f32.  M must be a multiple of 16 (100000 = 6250*16).
//
// Fragment layouts (ISA 7.12.2, 32-bit):
//   A 16x4:  lanes 0-15 -> M=lane,    v0=K0, v1=K1 ; lanes 16-31 -> v0=K2, v1=K3
//   B 4x16:  lanes 0-15 -> N=lane,    v0=K0, v1=K1 ; lanes 16-31 -> v0=K2, v1=K3
//   C 16x16: lanes 0-15 -> N=lane, vr -> M=r ; lanes 16-31 -> vr -> M=r+8
template <int K, int F>
__global__ __launch_bounds__(256) void k_gemm_wmma(const float* __restrict__ A,
                                                   const float* __restrict__ B,
                                                   float* __restrict__ C, int M) {
  constexpr int TF = F / 16;
  const int wave   = (int)((blockIdx.x * blockDim.x + threadIdx.x) >> 5);
  const int lane   = threadIdx.x & 31;
  const int tile_m = wave / TF;
  const int tile_f = wave % TF;
  if (tile_m * 16 >= M) return;          // wave-uniform: EXEC stays all-ones

  const int half = lane >> 4;            // 0: K pair {0,1}, 1: K pair {2,3}
  const int l    = lane & 15;
  const float* Arow = A + (size_t)(tile_m * 16 + l) * K + 2 * half;
  const int col     = tile_f * 16 + l;

  v8f c = {};
#pragma unroll
  for (int k0 = 0; k0 < K; k0 += 4) {
    v2f a = *(const v2f*)(Arow + k0);                  // global_load_b64
    v2f b;
    b.x = B[(k0 + 2 * half) * F + col];                // W is L2/L0 resident
    b.y = B[(k0 + 2 * half + 1) * F + col];
    c = __builtin_amdgcn_wmma_f32_16x16x4_f32(
        /*neg_a=*/false, a, /*neg_b=*/false, b,
        /*c_mod=*/(short)0, c, /*reuse_a=*/false, /*reuse_b=*/false);
  }

  float* Crow = C + (size_t)(tile_m * 16 + 8 * half) * F + col;
#pragma unroll
  for (int r = 0; r < 8; ++r)            // lanes 0-15 write 64B rows, coalesced
    Crow[(size_t)r * F] = c[r];
}

// ------------------------------------------------------ edge aggregation ---
// Self-loop term initializes agg: agg[n,f] = h[n,f] * dis[n]^2
__global__ void k_selfloop(const float* __restrict__ h, const float* __restrict__ dis,
                           float* __restrict__ agg, int total) {
  int i = blockIdx.x * blockDim.x + threadIdx.x;
  if (i >= total) return;
  float di = dis[i >> 6];
  agg[i] = h[i] * di * di;
}

// agg[dst,f] += h[src,f] * dis[src]*dis[dst]; thread = (edge, feat) so the
// 64-float row gather/scatter is fully coalesced across two waves.
__global__ void k_edge(const int* __restrict__ src, const int* __restrict__ dst,
                       const float* __restrict__ dis, const float* __restrict__ h,
                       float* __restrict__ agg, int total) {
  int i = blockIdx.x * blockDim.x + threadIdx.x;
  if (i >= total) return;
  int e = i >> 6;
  int f = i & 63;
  int s = src[e], d = dst[e];
  float norm = dis[s] * dis[d];
  atomicAdd(&agg[(size_t)d * FDIM + f], h[(size_t)s * FDIM + f] * norm);
}

// in place: a += bias, optional ReLU
__global__ void k_bias_act(float* __restrict__ a, const float* __restrict__ b,
                           int total, int do_relu) {
  int i = blockIdx.x * blockDim.x + threadIdx.x;
  if (i >= total) return;
  float v = a[i] + b[i & 63];
  if (do_relu) v = fmaxf(v, 0.0f);
  a[i] = v;
}

// ----------------------------------------------------------- scatter mean --
__global__ void k_cnt(const int* __restrict__ uidx, float* __restrict__ cnt, int N) {
  int n = blockIdx.x * blockDim.x + threadIdx.x;
  if (n < N) atomicAdd(&cnt[uidx[n]], 1.0f);
}

__global__ void k_scatter(const int* __restrict__ uidx, const float* __restrict__ h,
                          float* __restrict__ out, int total) {
  int i = blockIdx.x * blockDim.x + threadIdx.x;
  if (i >= total) return;
  int n = i >> 6;
  int f = i & 63;
  atomicAdd(&out[(size_t)uidx[n] * FDIM + f], h[i]);
}

__global__ void k_div(float* __restrict__ out, const float* __restrict__ cnt, int total) {
  int i = blockIdx.x * blockDim.x + threadIdx.x;
  if (i >= total) return;
  float c = cnt[i >> 6];
  out[i] = (c > 0.0f) ? out[i] / c : 0.0f;
}

// ------------------------------------------------------------------ launch --
extern "C" void kernel_launch(void* const* d_in, const int* in_sizes, int n_in,
                              void* d_out, int out_size, void* d_ws, size_t ws_size,
                              hipStream_t stream) {
  const float* x    = (const float*)d_in[0];
  const int*   eix  = (const int*)d_in[1];
  const int*   uidx = (const int*)d_in[2];
  const float* W1   = (const float*)d_in[3];
  const float* b1   = (const float*)d_in[4];
  const float* W2   = (const float*)d_in[5];
  const float* b2   = (const float*)d_in[6];
  float*       out  = (float*)d_out;

  const int IN = 128;
  const int N  = in_sizes[0] / IN;    // 100000
  const int E  = in_sizes[1] / 2;     // 1600000
  const int* src = eix;               // edge_index[0]
  const int* dst = eix + E;           // edge_index[1]

  // workspace layout (~52 MB): bufA | bufB | dis | cnt
  float* bufA = (float*)d_ws;                    // N*64 f32
  float* bufB = bufA + (size_t)N * FDIM;         // N*64 f32
  float* dis  = bufB + (size_t)N * FDIM;         // N f32 (deg, then deg^-1/2)
  float* cnt  = dis + N;                         // N f32

  const int B256 = 256;
  const int totNF = N * FDIM;                    // 6.4M
  const int totEF = E * FDIM;                    // 102.4M (fits int32)
  const int gemm_blocks = cdiv((long long)cdiv(N, 16) * (FDIM / 16) * 32, B256);

  // --- degree / normalization ---
  k_fill<<<cdiv(N, B256), B256, 0, stream>>>(dis, 1.0f, N);    // self loop
  k_deg<<<cdiv(E, B256), B256, 0, stream>>>(dst, dis, E);
  k_rsqrt_inplace<<<cdiv(N, B256), B256, 0, stream>>>(dis, N);

  // --- layer 1: h1 = x @ W1 ; agg ; +b1 ; relu ---
  k_gemm_wmma<128, FDIM><<<gemm_blocks, B256, 0, stream>>>(x, W1, bufA, N);
  k_selfloop<<<cdiv(totNF, B256), B256, 0, stream>>>(bufA, dis, bufB, totNF);
  k_edge<<<cdiv(totEF, B256), B256, 0, stream>>>(src, dst, dis, bufA, bufB, totEF);
  k_bias_act<<<cdiv(totNF, B256), B256, 0, stream>>>(bufB, b1, totNF, 1);

  // --- layer 2: h2 = relu_out @ W2 ; agg ; +b2 ---
  k_gemm_wmma<64, FDIM><<<gemm_blocks, B256, 0, stream>>>(bufB, W2, bufA, N);
  k_selfloop<<<cdiv(totNF, B256), B256, 0, stream>>>(bufA, dis, bufB, totNF);
  k_edge<<<cdiv(totEF, B256), B256, 0, stream>>>(src, dst, dis, bufA, bufB, totEF);
  k_bias_act<<<cdiv(totNF, B256), B256, 0, stream>>>(bufB, b2, totNF, 0);

  // --- scatter mean over user_idx (accumulate directly into d_out) ---
  k_fill<<<cdiv(totNF, B256), B256, 0, stream>>>(out, 0.0f, totNF);
  k_fill<<<cdiv(N, B256), B256, 0, stream>>>(cnt, 0.0f, N);
  k_cnt<<<cdiv(N, B256), B256, 0, stream>>>(uidx, cnt, N);
  k_scatter<<<cdiv(totNF, B256), B256, 0, stream>>>(uidx, bufB, out, totNF);
  k_div<<<cdiv(totNF, B256), B256, 0, stream>>>(out, cnt, totNF);
}